// KMeansCluster_30288109371843
// MI455X (gfx1250) — compile-verified
//
#include <hip/hip_runtime.h>
#include <hip/hip_bf16.h>

typedef __attribute__((ext_vector_type(2))) float v2f;
typedef __attribute__((ext_vector_type(8))) float v8f;

constexpr int N = 65536;
constexpr int D = 256;
constexpr int K = 512;
constexpr int EPOCHS = 10;
constexpr int PITCH = 260;                      // floats; 260 % 64 == 4 -> conflict-free fragment reads
constexpr int SMEM_BYTES = (64 + 16) * PITCH * 4; // 83,200 B

// ---------------------------------------------------------------------------
// Fused GEMM (x @ cent^T) + row argmin, using V_WMMA_F32_16X16X4_F32.
// Block: 128 threads = 4 waves. Block tile: 64 rows x all 512 centroids.
// Wave w owns rows [blockRow + 16w, +16) and sweeps all 32 column tiles.
// ---------------------------------------------------------------------------
__global__ __launch_bounds__(128) void assign_kernel(const float* __restrict__ x,
                                                     const float* __restrict__ cent,
                                                     int* __restrict__ idx) {
    extern __shared__ float smem[];
    float* lx = smem;                // 64 x PITCH  (X tile)
    float* lb = smem + 64 * PITCH;   // 16 x PITCH  (centroid tile)

    const int tid  = threadIdx.x;
    const int wave = tid >> 5;
    const int lane = tid & 31;
    const int lrow = lane & 15;      // N column within tile / M row for A-frag
    const int hi   = lane >> 4;      // half-wave select
    const int mb   = blockIdx.x * 64;

    // Stage X tile: 64 rows x 256 floats (float4 vectorized).
    for (int t = tid; t < 64 * 64; t += 128) {
        const int row = t >> 6;
        const int c4  = t & 63;
        const float4 v = ((const float4*)(x + (size_t)(mb + row) * D))[c4];
        float* p = &lx[row * PITCH + c4 * 4];
        p[0] = v.x; p[1] = v.y; p[2] = v.z; p[3] = v.w;
    }

    float bestv[8];
    int   besti[8];
#pragma unroll
    for (int r = 0; r < 8; ++r) { bestv[r] = __builtin_inff(); besti[r] = 0; }

    // A fragment: lane -> row (wave*16 + lrow), K = db + 2*hi + {0,1}
    // B fragment: lane -> centroid row lrow,    K = db + 2*hi + {0,1}
    const float* arow = &lx[(wave * 16 + lrow) * PITCH + 2 * hi];
    const float* brow = &lb[lrow * PITCH + 2 * hi];

    for (int nb = 0; nb < K; nb += 16) {
        __syncthreads();   // covers X-tile load on first pass; tile reuse after
        for (int t = tid; t < 16 * 64; t += 128) {
            const int row = t >> 6;
            const int c4  = t & 63;
            const float4 v = ((const float4*)(cent + (size_t)(nb + row) * D))[c4];
            float* p = &lb[row * PITCH + c4 * 4];
            p[0] = v.x; p[1] = v.y; p[2] = v.z; p[3] = v.w;
        }
        __syncthreads();

        v8f a0 = {0.f,0.f,0.f,0.f,0.f,0.f,0.f,0.f};
        v8f a1 = a0, a2 = a0, a3 = a0;
#pragma unroll
        for (int db = 0; db < D; db += 16) {
            v2f xa0 = *(const v2f*)(arow + db + 0);
            v2f xb0 = *(const v2f*)(brow + db + 0);
            a0 = __builtin_amdgcn_wmma_f32_16x16x4_f32(false, xa0, false, xb0, (short)0, a0, false, false);
            v2f xa1 = *(const v2f*)(arow + db + 4);
            v2f xb1 = *(const v2f*)(brow + db + 4);
            a1 = __builtin_amdgcn_wmma_f32_16x16x4_f32(false, xa1, false, xb1, (short)0, a1, false, false);
            v2f xa2 = *(const v2f*)(arow + db + 8);
            v2f xb2 = *(const v2f*)(brow + db + 8);
            a2 = __builtin_amdgcn_wmma_f32_16x16x4_f32(false, xa2, false, xb2, (short)0, a2, false, false);
            v2f xa3 = *(const v2f*)(arow + db + 12);
            v2f xb3 = *(const v2f*)(brow + db + 12);
            a3 = __builtin_amdgcn_wmma_f32_16x16x4_f32(false, xa3, false, xb3, (short)0, a3, false, false);
        }
        v8f acc = (a0 + a1) + (a2 + a3);

        // C/D layout: vgpr r, lane -> element (M = r + 8*hi, N = lrow)
        const int col = nb + lrow;
#pragma unroll
        for (int r = 0; r < 8; ++r) {
            const float s = acc[r];
            if (s < bestv[r]) { bestv[r] = s; besti[r] = col; }   // ascending nb => first-occurrence ties
        }
    }

    // Argmin across the 16 lanes of each half-wave (masks 1..8 stay in-half).
#pragma unroll
    for (int r = 0; r < 8; ++r) {
        float bv = bestv[r];
        int   bi = besti[r];
#pragma unroll
        for (int m = 1; m <= 8; m <<= 1) {
            const float ov = __shfl_xor(bv, m, 32);
            const int   oi = __shfl_xor(bi, m, 32);
            if (ov < bv || (ov == bv && oi < bi)) { bv = ov; bi = oi; }
        }
        bestv[r] = bv; besti[r] = bi;
    }
    if (lrow == 0) {
#pragma unroll
        for (int r = 0; r < 8; ++r)
            idx[mb + wave * 16 + r + 8 * hi] = besti[r];
    }
}

// ---------------------------------------------------------------------------
// cent[0:K] = x[0:K]  (initial centroids)
// ---------------------------------------------------------------------------
__global__ void init_cent_kernel(const float* __restrict__ x, float* __restrict__ cent) {
    const int t = blockIdx.x * blockDim.x + threadIdx.x;     // float4 index
    ((float4*)cent)[t] = ((const float4*)x)[t];
}

// ---------------------------------------------------------------------------
// Zero sums (K*D) and counts (K) — contiguous in workspace.
// ---------------------------------------------------------------------------
__global__ void clear_kernel(float* __restrict__ p, int n) {
    const int t = blockIdx.x * blockDim.x + threadIdx.x;
    if (t < n) p[t] = 0.0f;
}

// ---------------------------------------------------------------------------
// Segment sums: one wave per point; 8 float atomics per lane + 1 count atomic.
// ---------------------------------------------------------------------------
__global__ __launch_bounds__(256) void accum_kernel(const float* __restrict__ x,
                                                    const int* __restrict__ idx,
                                                    float* __restrict__ sums,
                                                    float* __restrict__ counts) {
    const int point = (blockIdx.x * blockDim.x + threadIdx.x) >> 5;
    const int lane  = threadIdx.x & 31;
    if (point >= N) return;
    const int k = idx[point];
    const float* xr = x + (size_t)point * D;
    float* sr = sums + (size_t)k * D;
#pragma unroll
    for (int j = 0; j < D / 32; ++j)
        atomicAdd(&sr[lane + 32 * j], xr[lane + 32 * j]);
    if (lane == 0) atomicAdd(&counts[k], 1.0f);
}

// ---------------------------------------------------------------------------
// cent = counts>0 ? sums/max(counts,1) : cent  (matches jnp.where semantics)
// ---------------------------------------------------------------------------
__global__ void update_kernel(const float* __restrict__ sums,
                              const float* __restrict__ counts,
                              float* __restrict__ cent) {
    const int t = blockIdx.x * blockDim.x + threadIdx.x;     // 0..K*D-1
    const int k = t >> 8;                                    // D == 256
    const float cnt = counts[k];
    const float safe = fmaxf(cnt, 1.0f);
    const float v = sums[t] / safe;
    if (cnt > 0.0f) cent[t] = v;
}

extern "C" void kernel_launch(void* const* d_in, const int* in_sizes, int n_in,
                              void* d_out, int out_size, void* d_ws, size_t ws_size,
                              hipStream_t stream) {
    (void)in_sizes; (void)n_in; (void)out_size; (void)ws_size;

    const float* x = (const float*)d_in[0];
    int* idx = (int*)d_out;

    float* ws     = (float*)d_ws;
    float* cent   = ws;                 // K*D floats
    float* sums   = ws + (size_t)K * D; // K*D floats
    float* counts = ws + (size_t)2 * K * D; // K floats (contiguous after sums)

    // cent0 = x[:K]  (K*D/4 float4 elements, 256 threads/block)
    init_cent_kernel<<<(K * D / 4) / 256, 256, 0, stream>>>(x, cent);

    for (int e = 0; e < EPOCHS; ++e) {
        assign_kernel<<<N / 64, 128, SMEM_BYTES, stream>>>(x, cent, idx);
        if (e == EPOCHS - 1) break;     // final epoch's idx is the output; no update needed
        const int nclr = K * D + K;
        clear_kernel<<<(nclr + 255) / 256, 256, 0, stream>>>(sums, nclr);
        accum_kernel<<<N * 32 / 256, 256, 0, stream>>>(x, idx, sums, counts);
        update_kernel<<<(K * D) / 256, 256, 0, stream>>>(sums, counts, cent);
    }
}